// DenseCapRoIHeads_4209067950455
// MI455X (gfx1250) — compile-verified
//
#include <hip/hip_runtime.h>
#include <stdint.h>

// ---------------- problem constants (from reference) ----------------
#define B_IMG   8
#define NPROP   8000
#define NGT     256
#define NALL    (NPROP + NGT)         // 8256
#define PRE_NMS 1000
#define DETS    300
#define SORT_N  16384                 // next pow2 >= NALL
#define IMG_WH  800.0f
#define SCORE_T 0.05f
#define NMS_T   0.5f
#define FGBG_T  0.5f                  // FG_T == BG_T == 0.5
#define DW_CLIP 4.135166556742356f    // log(1000/16)

// output layout (flat concat of tuple, batch-major)
#define OUT_LABELS 0
#define OUT_REG    (B_IMG * NALL)                  // 66048
#define OUT_DETB   (OUT_REG + B_IMG * NALL * 4)    // 330240
#define OUT_DETS   (OUT_DETB + B_IMG * DETS * 4)   // 339840

// ---------------- CDNA5 async global->LDS helpers ----------------
__device__ __forceinline__ void async_g2l_b128(void* lds_dst, const void* gsrc) {
  // LDS destination address is taken from the VDST VGPR; low 32 bits of a generic LDS pointer are
  // the workgroup-relative LDS byte offset.
  uint32_t lo = (uint32_t)(uintptr_t)lds_dst;
  asm volatile("global_load_async_to_lds_b128 %0, %1, off"
               :: "v"(lo), "v"(gsrc) : "memory");
}
__device__ __forceinline__ void wait_asynccnt0() {
  asm volatile("s_wait_asynccnt 0" ::: "memory");
}

// ============================================================================
// Kernel 1: per-proposal IoU-vs-GT max/argmax -> labels + encode targets
// grid = (ceil(NALL/256), B), block = 256
// ============================================================================
__global__ __launch_bounds__(256) void k_label_encode(
    const float* __restrict__ proposals,   // (B, NPROP, 4)
    const float* __restrict__ gt_boxes,    // (B, NGT, 4)
    float* __restrict__ out) {
  __shared__ __align__(16) float s_gt[NGT * 4];   // 4 KB

  const int b   = blockIdx.y;
  const int tid = threadIdx.x;
  const float* gt_img = gt_boxes + (size_t)b * NGT * 4;

  // async-stage GT boxes: 256 lanes x 16B = 4KB
  async_g2l_b128(&s_gt[tid * 4], &gt_img[tid * 4]);
  wait_asynccnt0();
  __syncthreads();

  const int p = blockIdx.x * 256 + tid;
  if (p >= NALL) return;

  // prop_all[p] = concat(proposals, gt)[p]
  float p0, p1, p2, p3;
  if (p < NPROP) {
    const float* pp = proposals + ((size_t)b * NPROP + p) * 4;
    p0 = pp[0]; p1 = pp[1]; p2 = pp[2]; p3 = pp[3];
  } else {
    const int g = (p - NPROP) * 4;
    p0 = s_gt[g]; p1 = s_gt[g + 1]; p2 = s_gt[g + 2]; p3 = s_gt[g + 3];
  }
  const float area_p = (p2 - p0) * (p3 - p1);

  // max / argmax over 256 GT (strict > keeps first max, matching jnp.argmax)
  float best = -1.0f;
  int   bi   = 0;
  for (int i = 0; i < NGT; ++i) {
    const float g0 = s_gt[i * 4 + 0], g1 = s_gt[i * 4 + 1];
    const float g2 = s_gt[i * 4 + 2], g3 = s_gt[i * 4 + 3];
    const float area_g = (g2 - g0) * (g3 - g1);
    float w = fminf(g2, p2) - fmaxf(g0, p0);
    float h = fminf(g3, p3) - fmaxf(g1, p1);
    w = fmaxf(w, 0.0f); h = fmaxf(h, 0.0f);
    const float inter = w * h;
    const float iou = inter / (area_g + area_p - inter);
    if (iou > best) { best = iou; bi = i; }
  }

  // labels: vals<BG_T -> 0 ; elif vals<FG_T -> -1 (dead, thresholds equal) ; else 1
  float label = (best < FGBG_T) ? 0.0f : ((best < FGBG_T) ? -1.0f : 1.0f);
  out[OUT_LABELS + (size_t)b * NALL + p] = label;

  // encode(prop_all, gt[argmax])
  const float g0 = s_gt[bi * 4 + 0], g1 = s_gt[bi * 4 + 1];
  const float g2 = s_gt[bi * 4 + 2], g3 = s_gt[bi * 4 + 3];
  const float pw = p2 - p0, ph = p3 - p1;
  const float pcx = p0 + 0.5f * pw, pcy = p1 + 0.5f * ph;
  const float gw = g2 - g0, gh = g3 - g1;
  const float gcx = g0 + 0.5f * gw, gcy = g1 + 0.5f * gh;
  float* rt = out + OUT_REG + ((size_t)b * NALL + p) * 4;
  rt[0] = 10.0f * (gcx - pcx) / pw;
  rt[1] = 10.0f * (gcy - pcy) / ph;
  rt[2] = 5.0f * logf(gw / pw);
  rt[3] = 5.0f * logf(gh / ph);
}

// ============================================================================
// Kernel 2: per-image scores -> exact bitonic top-1000 -> bitmask NMS ->
//           stable top-300 fill.  grid = B, block = 1024 (32 waves)
// ============================================================================
__global__ __launch_bounds__(1024) void k_sort_nms(
    const float* __restrict__ proposals,
    const float* __restrict__ gt_boxes,
    const float* __restrict__ logits,      // (B, NALL, 2)
    const float* __restrict__ breg,        // (B, NALL, 8)
    float* __restrict__ out) {
  union SortMask {
    unsigned long long keys[SORT_N];       // 131072 B (sort phase)
    unsigned int       mask[PRE_NMS * 32]; // 128000 B (NMS phase, reuses LDS)
  };
  __shared__ SortMask sm;
  __shared__ float s_tb[PRE_NMS][4];       // top-1000 boxes   (16000 B)
  __shared__ float s_ts[PRE_NMS];          // top-1000 scores  ( 4000 B)
  __shared__ int   s_scan[1024];           // block scan       ( 4096 B)
  __shared__ unsigned int s_keep[32];      // NMS keep bitmask

  const int b   = blockIdx.x;
  const int tid = threadIdx.x;

  // ---- Phase A: build 64-bit sort keys -------------------------------------
  for (int p = tid; p < SORT_N; p += 1024) {
    unsigned long long key = 0ull;  // padding sorts below every real entry
    if (p < NALL) {
      if (p + 1024 < NALL)
        __builtin_prefetch(&logits[((size_t)b * NALL + p + 1024) * 2], 0, 3);
      const float l0 = logits[((size_t)b * NALL + p) * 2 + 0];
      const float l1 = logits[((size_t)b * NALL + p) * 2 + 1];
      const float m  = fmaxf(l0, l1);
      const float e0 = expf(l0 - m), e1 = expf(l1 - m);
      const float s  = e1 / (e0 + e1);                 // softmax[:,1]
      const float sc = (s > SCORE_T) ? s : -1.0f;
      unsigned int u = __float_as_uint(sc);
      u = (u & 0x80000000u) ? ~u : (u | 0x80000000u);  // order-preserving map
      // low word ~p => on equal score, SMALLER index sorts FIRST (stable top_k)
      key = ((unsigned long long)u << 32) |
            (unsigned long long)(0xFFFFFFFFu - (unsigned)p);
    }
    sm.keys[p] = key;
  }
  __syncthreads();

  // ---- Phase B: bitonic sort, descending -----------------------------------
  for (int k = 2; k <= SORT_N; k <<= 1) {
    for (int j = k >> 1; j > 0; j >>= 1) {
      for (int i = tid; i < SORT_N; i += 1024) {
        const int ixj = i ^ j;
        if (ixj > i) {
          const unsigned long long a = sm.keys[i];
          const unsigned long long c = sm.keys[ixj];
          const bool desc = ((i & k) == 0);
          if (desc ? (a < c) : (a > c)) { sm.keys[i] = c; sm.keys[ixj] = a; }
        }
      }
      __syncthreads();
    }
  }

  // ---- Phase C: gather top-1000, decode + clip class-1 boxes ---------------
  if (tid < PRE_NMS) {
    const unsigned long long key = sm.keys[tid];
    unsigned int hi = (unsigned int)(key >> 32);
    hi = (hi & 0x80000000u) ? (hi ^ 0x80000000u) : ~hi;  // invert order map
    const float sc = __uint_as_float(hi);
    const int idx = (int)(0xFFFFFFFFu - (unsigned int)key);

    float p0, p1, p2, p3;
    if (idx < NPROP) {
      const float* pp = proposals + ((size_t)b * NPROP + idx) * 4;
      p0 = pp[0]; p1 = pp[1]; p2 = pp[2]; p3 = pp[3];
    } else {
      const float* gp = gt_boxes + ((size_t)b * NGT + (idx - NPROP)) * 4;
      p0 = gp[0]; p1 = gp[1]; p2 = gp[2]; p3 = gp[3];
    }
    const float* r = breg + ((size_t)b * NALL + idx) * 8 + 4;  // class-1 deltas
    const float pw = p2 - p0, ph = p3 - p1;
    const float pcx = p0 + 0.5f * pw, pcy = p1 + 0.5f * ph;
    const float dx = r[0] / 10.0f, dy = r[1] / 10.0f;
    const float dw = fminf(r[2] / 5.0f, DW_CLIP);
    const float dh = fminf(r[3] / 5.0f, DW_CLIP);
    const float cx = dx * pw + pcx, cy = dy * ph + pcy;
    const float w  = expf(dw) * pw, h = expf(dh) * ph;
    s_tb[tid][0] = fminf(fmaxf(cx - 0.5f * w, 0.0f), IMG_WH);
    s_tb[tid][1] = fminf(fmaxf(cy - 0.5f * h, 0.0f), IMG_WH);
    s_tb[tid][2] = fminf(fmaxf(cx + 0.5f * w, 0.0f), IMG_WH);
    s_tb[tid][3] = fminf(fmaxf(cy + 0.5f * h, 0.0f), IMG_WH);
    s_ts[tid] = sc;
  }
  __syncthreads();   // keys dead beyond this point; LDS region becomes mask

  // ---- Phase D: 1000x1000 suppression bitmask (j>i and IoU>0.5) ------------
  for (int wrd = tid; wrd < PRE_NMS * 32; wrd += 1024) {
    const int i  = wrd >> 5;
    const int wc = wrd & 31;
    const float a0 = s_tb[i][0], a1 = s_tb[i][1], a2 = s_tb[i][2], a3 = s_tb[i][3];
    const float area_i = (a2 - a0) * (a3 - a1);
    unsigned int bits = 0u;
    const int jbase = wc << 5;
    for (int bj = 0; bj < 32; ++bj) {
      const int j = jbase + bj;
      if (j < PRE_NMS && j > i) {
        const float c0 = s_tb[j][0], c1 = s_tb[j][1];
        const float c2 = s_tb[j][2], c3 = s_tb[j][3];
        const float area_j = (c2 - c0) * (c3 - c1);
        const float iw = fmaxf(fminf(a2, c2) - fmaxf(a0, c0), 0.0f);
        const float ih = fmaxf(fminf(a3, c3) - fmaxf(a1, c1), 0.0f);
        const float inter = iw * ih;
        const float iou = inter / (area_i + area_j - inter);  // NaN>T == false
        if (iou > NMS_T) bits |= (1u << bj);
      }
    }
    sm.mask[wrd] = bits;
  }
  __syncthreads();

  // ---- Phase E: greedy NMS scan in ONE wave32 (no barriers) ----------------
  if (tid < 32) {
    unsigned int keep = (tid < 31) ? 0xFFFFFFFFu : 0x000000FFu;  // j<1000
    for (int i = 0; i < PRE_NMS; ++i) {
      const unsigned int kw  = __shfl(keep, i >> 5, 32);
      const unsigned int row = sm.mask[(i << 5) + tid];
      const unsigned int bit = (kw >> (i & 31)) & 1u;
      keep &= ~(bit ? row : 0u);   // branchless: wave stays convergent
    }
    s_keep[tid] = keep;
  }
  __syncthreads();

  // ---- Phase F: stable top-300 fill ----------------------------------------
  float* detb = out + OUT_DETB + (size_t)b * DETS * 4;
  float* dets = out + OUT_DETS + (size_t)b * DETS;

  int flag = 0;
  if (tid < PRE_NMS) {
    const unsigned int kb = (s_keep[tid >> 5] >> (tid & 31)) & 1u;
    flag = (kb && (s_ts[tid] > 0.0f)) ? 1 : 0;
  }

  // inclusive scan #1: kept entries
  s_scan[tid] = flag;
  __syncthreads();
  for (int off = 1; off < 1024; off <<= 1) {
    const int add = (tid >= off) ? s_scan[tid - off] : 0;
    __syncthreads();
    s_scan[tid] += add;
    __syncthreads();
  }
  const int incl1  = s_scan[tid];
  const int n_kept = s_scan[1023];
  __syncthreads();

  if (flag) {
    const int rank = incl1 - 1;                 // already score-descending
    if (rank < DETS) {
      dets[rank] = s_ts[tid];
      detb[rank * 4 + 0] = s_tb[tid][0];
      detb[rank * 4 + 1] = s_tb[tid][1];
      detb[rank * 4 + 2] = s_tb[tid][2];
      detb[rank * 4 + 3] = s_tb[tid][3];
    }
  }

  // inclusive scan #2: the -1 entries fill remaining slots in index order
  const int nflag = (tid < PRE_NMS) ? (1 - flag) : 0;
  s_scan[tid] = nflag;
  __syncthreads();
  for (int off = 1; off < 1024; off <<= 1) {
    const int add = (tid >= off) ? s_scan[tid - off] : 0;
    __syncthreads();
    s_scan[tid] += add;
    __syncthreads();
  }
  const int incl2 = s_scan[tid];

  if (nflag) {
    const int slot = n_kept + incl2 - 1;
    if (slot < DETS) {
      dets[slot] = -1.0f;
      detb[slot * 4 + 0] = s_tb[tid][0];
      detb[slot * 4 + 1] = s_tb[tid][1];
      detb[slot * 4 + 2] = s_tb[tid][2];
      detb[slot * 4 + 3] = s_tb[tid][3];
    }
  }
}

// ============================================================================
extern "C" void kernel_launch(void* const* d_in, const int* in_sizes, int n_in,
                              void* d_out, int out_size, void* d_ws, size_t ws_size,
                              hipStream_t stream) {
  (void)in_sizes; (void)n_in; (void)out_size; (void)d_ws; (void)ws_size;
  const float* proposals = (const float*)d_in[0];
  const float* gt_boxes  = (const float*)d_in[1];
  const float* logits    = (const float*)d_in[2];
  const float* breg      = (const float*)d_in[3];
  float* out = (float*)d_out;

  dim3 g1((NALL + 255) / 256, B_IMG);
  k_label_encode<<<g1, dim3(256), 0, stream>>>(proposals, gt_boxes, out);
  k_sort_nms<<<dim3(B_IMG), dim3(1024), 0, stream>>>(proposals, gt_boxes,
                                                     logits, breg, out);
}